// variance_outlier_extraction_layer_40870908789394
// MI455X (gfx1250) — compile-verified
//
#include <hip/hip_runtime.h>
#include <hip/hip_bf16.h>

typedef __attribute__((ext_vector_type(2))) float v2f;
typedef __attribute__((ext_vector_type(8))) float v8f;

#define BDIM 8
#define NDIM 2048
#define MDIM 2048

// ---------------------------------------------------------------------------
// Kernel 1: per-row sum and sum-of-squares via V_WMMA_F32_16X16X4_F32.
// One block = one (b, 16-row strip). 256 threads = 8 waves.
// LDS-staged 16x256 tiles, coalesced global loads, conflict-free fragment
// reads (row stride padded to 260 dwords). B matrix = all ones, so
// D[i][j] = sum_k A[i][k] (replicated over j) in full f32 precision.
// ---------------------------------------------------------------------------
#define CHUNK 256
#define PAD 260

__global__ __launch_bounds__(256) void loo_rowstats_kernel(
    const float* __restrict__ x, float* __restrict__ s_r, float* __restrict__ ss_r) {
  __shared__ float tile[16 * PAD];
  __shared__ float red_s[8][16];
  __shared__ float red_ss[8][16];

  const int t = threadIdx.x;
  const int wave = t >> 5;
  const int lane = t & 31;
  const int bi = blockIdx.x;
  const int b = bi / (NDIM / 16);
  const int r0 = (bi % (NDIM / 16)) * 16;
  const float* __restrict__ base =
      x + (size_t)b * NDIM * MDIM + (size_t)r0 * MDIM;

  v2f ones;
  ones.x = 1.0f;
  ones.y = 1.0f;
  v8f acc_s = {};
  v8f acc_ss = {};

  const int row_l = lane & 15;          // A-matrix M index for this lane
  const int colsel = (lane >> 4) << 1;  // K offset: 0 (lanes 0-15) or 2 (16-31)
  const float* __restrict__ trow = &tile[row_l * PAD];

  for (int c0 = 0; c0 < MDIM; c0 += CHUNK) {
    // Coalesced load of 16 x 256 tile into LDS.
#pragma unroll
    for (int i = 0; i < 16; ++i) {
      tile[i * PAD + t] = base[(size_t)i * MDIM + (c0 + t)];
    }
    __syncthreads();

    // Each wave reduces its 32-column slice of the chunk: 8 K-steps of 4.
#pragma unroll
    for (int kk = 0; kk < 8; ++kk) {
      const int k0 = wave * 32 + kk * 4 + colsel;
      v2f a;
      a.x = trow[k0];
      a.y = trow[k0 + 1];
      acc_s = __builtin_amdgcn_wmma_f32_16x16x4_f32(
          false, a, false, ones, (short)0, acc_s, false, false);
      v2f a2;
      a2.x = a.x * a.x;
      a2.y = a.y * a.y;
      acc_ss = __builtin_amdgcn_wmma_f32_16x16x4_f32(
          false, a2, false, ones, (short)0, acc_ss, false, false);
    }
    __syncthreads();
  }

  // D layout: VGPR r, lane 0 -> row r; lane 16 -> row 8+r. Spill per-wave
  // row sums to LDS, then 16 threads combine the 8 waves in fixed order.
  if (lane == 0) {
#pragma unroll
    for (int r = 0; r < 8; ++r) {
      red_s[wave][r] = acc_s[r];
      red_ss[wave][r] = acc_ss[r];
    }
  } else if (lane == 16) {
#pragma unroll
    for (int r = 0; r < 8; ++r) {
      red_s[wave][8 + r] = acc_s[r];
      red_ss[wave][8 + r] = acc_ss[r];
    }
  }
  __syncthreads();

  if (t < 16) {
    float s = 0.0f, ss = 0.0f;
#pragma unroll
    for (int w = 0; w < 8; ++w) {
      s += red_s[w][t];
      ss += red_ss[w][t];
    }
    s_r[b * NDIM + r0 + t] = s;
    ss_r[b * NDIM + r0 + t] = ss;
  }
}

// ---------------------------------------------------------------------------
// Kernel 2: per-column sum and sum-of-squares. One block = (b, 64-col group),
// 256 threads = 64 cols x 4 row segments of 512 rows. Coalesced column walk
// (hits L2 after kernel 1), fixed-order segment combine => deterministic.
// ---------------------------------------------------------------------------
__global__ __launch_bounds__(256) void loo_colstats_kernel(
    const float* __restrict__ x, float* __restrict__ s_c, float* __restrict__ ss_c) {
  __shared__ float red_s[4][64];
  __shared__ float red_ss[4][64];

  const int t = threadIdx.x;
  const int col_l = t & 63;
  const int seg = t >> 6;
  const int bi = blockIdx.x;
  const int b = bi >> 5;   // 32 column groups per batch (2048/64)
  const int cg = bi & 31;
  const int col = cg * 64 + col_l;

  const float* __restrict__ base =
      x + (size_t)b * NDIM * MDIM + (size_t)(seg * (NDIM / 4)) * MDIM + col;

  float s = 0.0f, ss = 0.0f;
#pragma unroll 4
  for (int r = 0; r < NDIM / 4; ++r) {
    const float v = base[(size_t)r * MDIM];
    s += v;
    ss += v * v;
  }
  red_s[seg][col_l] = s;
  red_ss[seg][col_l] = ss;
  __syncthreads();

  if (t < 64) {
    float ts = 0.0f, tss = 0.0f;
#pragma unroll
    for (int g = 0; g < 4; ++g) {
      ts += red_s[g][t];
      tss += red_ss[g][t];
    }
    s_c[b * MDIM + cg * 64 + t] = ts;
    ss_c[b * MDIM + cg * 64 + t] = tss;
  }
}

// ---------------------------------------------------------------------------
// Kernel 3: elementwise LOO-variance min + row flip, float4 vectorized.
// ---------------------------------------------------------------------------
__device__ __forceinline__ float loo_var(float xv, float s, float ss, float inv) {
  const float x2 = xv * xv;
  const float mean = (s - xv) * inv;
  return (ss - x2) * inv - mean * mean;
}

__global__ __launch_bounds__(256) void loo_finalize_kernel(
    const float* __restrict__ x, const float* __restrict__ s_r,
    const float* __restrict__ ss_r, const float* __restrict__ s_c,
    const float* __restrict__ ss_c, float* __restrict__ out) {
  const size_t idx = (size_t)blockIdx.x * blockDim.x + threadIdx.x;  // float4 id
  const int m4 = (int)(idx & (MDIM / 4 - 1));  // column group (512 per row)
  const size_t nb = idx >> 9;
  const int n = (int)(nb & (NDIM - 1));
  const int b = (int)(nb >> 11);

  const float4 xv = ((const float4*)x)[idx];
  const float sr = s_r[b * NDIM + n];
  const float ssr = ss_r[b * NDIM + n];
  const float4 scv = ((const float4*)s_c)[(size_t)b * (MDIM / 4) + m4];
  const float4 sscv = ((const float4*)ss_c)[(size_t)b * (MDIM / 4) + m4];

  const float inv_m = 1.0f / (float)(MDIM - 1);
  const float inv_n = 1.0f / (float)(NDIM - 1);

  float4 o;
  o.x = fminf(loo_var(xv.x, sr, ssr, inv_m), loo_var(xv.x, scv.x, sscv.x, inv_n));
  o.y = fminf(loo_var(xv.y, sr, ssr, inv_m), loo_var(xv.y, scv.y, sscv.y, inv_n));
  o.z = fminf(loo_var(xv.z, sr, ssr, inv_m), loo_var(xv.z, scv.z, sscv.z, inv_n));
  o.w = fminf(loo_var(xv.w, sr, ssr, inv_m), loo_var(xv.w, scv.w, sscv.w, inv_n));

  const size_t oidx =
      ((size_t)b * NDIM + (size_t)(NDIM - 1 - n)) * (MDIM / 4) + m4;
  ((float4*)out)[oidx] = o;
}

extern "C" void kernel_launch(void* const* d_in, const int* in_sizes, int n_in,
                              void* d_out, int out_size, void* d_ws, size_t ws_size,
                              hipStream_t stream) {
  (void)in_sizes; (void)n_in; (void)out_size; (void)ws_size;
  const float* x = (const float*)d_in[0];
  float* out = (float*)d_out;

  // Workspace: 4 stats arrays of B*2048 floats each (256 KiB total).
  float* ws = (float*)d_ws;
  float* s_r = ws;                    // (B, N)
  float* ss_r = ws + BDIM * NDIM;     // (B, N)
  float* s_c = ws + 2 * BDIM * NDIM;  // (B, M)
  float* ss_c = ws + 3 * BDIM * NDIM; // (B, M)

  // Pass 1a: row stats (WMMA reduction). One block per 16-row strip.
  loo_rowstats_kernel<<<BDIM * (NDIM / 16), 256, 0, stream>>>(x, s_r, ss_r);
  // Pass 1b: column stats (reads served from L2).
  loo_colstats_kernel<<<BDIM * (MDIM / 64), 256, 0, stream>>>(x, s_c, ss_c);
  // Pass 2: elementwise combine + flip (reads served from L2).
  const size_t n4 = (size_t)BDIM * NDIM * (MDIM / 4);
  loo_finalize_kernel<<<(unsigned)(n4 / 256), 256, 0, stream>>>(
      x, s_r, ss_r, s_c, ss_c, out);
}